// DigitConvolutionalModel_25735444037858
// MI455X (gfx1250) — compile-verified
//
#include <hip/hip_runtime.h>

typedef __attribute__((ext_vector_type(2))) float f32x2;
typedef __attribute__((ext_vector_type(8))) float f32x8;

#define N_IMGS          65536
#define IMG_H           28
#define IMG_W           28
#define OH              26
#define FEAT            676     // 26*26
#define NOUT            10
#define WAVES_PER_BLOCK 2
#define IMGS_PER_WAVE   16
#define IMGS_PER_BLOCK  (WAVES_PER_BLOCK * IMGS_PER_WAVE)
#define BLOCK_THREADS   (WAVES_PER_BLOCK * 32)

__launch_bounds__(BLOCK_THREADS, 1)
__global__ void conv_relu_linear_wmma(const float* __restrict__ x,
                                      const float* __restrict__ cw,
                                      const float* __restrict__ Wlin,
                                      const float* __restrict__ bias,
                                      float* __restrict__ out) {
  // Zero-padded 676x16 copy of W (cols 10..15 = 0): 43,264 B
  __shared__ float Wlds[FEAT * 16];
  // Per-wave feature tile: 16 images x 676 f32: 43,264 B each
  __shared__ float feat[WAVES_PER_BLOCK][IMGS_PER_WAVE * FEAT];

  const int tid = threadIdx.x;

  // ---- Stage padded W into LDS (cooperative, once per block) ----
  for (int i = tid; i < FEAT * 16; i += BLOCK_THREADS) {
    const int c = i & 15;
    const int r = i >> 4;
    Wlds[i] = (c < NOUT) ? Wlin[r * NOUT + c] : 0.0f;
  }

  // ---- Conv weights (broadcast; scalar-cached) ----
  float wk[9];
  #pragma unroll
  for (int i = 0; i < 9; ++i) wk[i] = cw[i];

  const int lane = tid & 31;
  const int wave = tid >> 5;
  const int m    = lane & 15;   // image-in-wave (A rows) AND output column (B cols)
  const int half = lane >> 4;

  // ---- Conv + ReLU phase: lanes m and m+16 split image m's 26 rows ----
  const int img = blockIdx.x * IMGS_PER_BLOCK + wave * IMGS_PER_WAVE + m;
  const float* __restrict__ ximg = x + img * (IMG_H * IMG_W);
  float* __restrict__ fdst = &feat[wave][m * FEAT];

  const int rbeg = half * 13;
  #pragma unroll 1
  for (int r = rbeg; r < rbeg + 13; ++r) {
    // Prefetch the row that enters the window next iteration (speculative;
    // dropped silently if it runs past the mapped region on the last tile).
    __builtin_prefetch(ximg + (r + 3) * IMG_W, 0, 0);

    float rowv[3][IMG_W];
    #pragma unroll
    for (int rr = 0; rr < 3; ++rr) {
      #pragma unroll
      for (int c4 = 0; c4 < 7; ++c4) {
        const float4 v = *(const float4*)(ximg + (r + rr) * IMG_W + c4 * 4);
        rowv[rr][c4 * 4 + 0] = v.x;
        rowv[rr][c4 * 4 + 1] = v.y;
        rowv[rr][c4 * 4 + 2] = v.z;
        rowv[rr][c4 * 4 + 3] = v.w;
      }
    }
    float* __restrict__ drow = fdst + r * OH;   // 8B-aligned (104*r + 2704*m)
    #pragma unroll
    for (int c2 = 0; c2 < 13; ++c2) {
      f32x2 o;
      #pragma unroll
      for (int k = 0; k < 2; ++k) {
        const int c = 2 * c2 + k;
        float s = 0.0f;
        #pragma unroll
        for (int dr = 0; dr < 3; ++dr)
          #pragma unroll
          for (int dc = 0; dc < 3; ++dc)
            s = fmaf(rowv[dr][c + dc], wk[dr * 3 + dc], s);
        o[k] = fmaxf(s, 0.0f);
      }
      *(f32x2*)(drow + 2 * c2) = o;
    }
  }

  __syncthreads();

  // ---- WMMA phase: D(16x16) = feat(16x676) * Wpad(676x16) + bias ----
  // C/D layout: VGPR j -> M = j + 8*half (lanes 16-31), N = lane&15.
  const float bcol = (m < NOUT) ? bias[m] : 0.0f;
  f32x8 c;
  #pragma unroll
  for (int j = 0; j < 8; ++j) c[j] = bcol;

  // A fragment: lane (m, half) holds A[m][4k + 2*half + {0,1}]  (one b64 LDS load)
  const float* __restrict__ abase = &feat[wave][m * FEAT + 2 * half];
  // B fragment: lane (n=m, half) holds B[4k + 2*half + {0,1}][m]
  const float* __restrict__ bbase = &Wlds[2 * half * 16 + m];

  // Unroll x4: batch DS loads so s_wait_dscnt waits on partial counts instead
  // of fully serializing load->wmma each iteration (169 % 4 handled below).
  #pragma unroll 4
  for (int k = 0; k < 168; ++k) {
    const f32x2 a = *(const f32x2*)(abase + 4 * k);  // 8B-aligned
    f32x2 bb;
    bb.x = bbase[k * 64];
    bb.y = bbase[k * 64 + 16];
    c = __builtin_amdgcn_wmma_f32_16x16x4_f32(false, a, false, bb,
                                              (short)0, c, false, false);
  }
  {
    const int k = 168;
    const f32x2 a = *(const f32x2*)(abase + 4 * k);
    f32x2 bb;
    bb.x = bbase[k * 64];
    bb.y = bbase[k * 64 + 16];
    c = __builtin_amdgcn_wmma_f32_16x16x4_f32(false, a, false, bb,
                                              (short)0, c, false, false);
  }

  // ---- Store 16x10 tile ----
  const int orow0 = blockIdx.x * IMGS_PER_BLOCK + wave * IMGS_PER_WAVE + half * 8;
  if (m < NOUT) {
    #pragma unroll
    for (int j = 0; j < 8; ++j) {
      out[(orow0 + j) * NOUT + m] = c[j];
    }
  }
}

extern "C" void kernel_launch(void* const* d_in, const int* in_sizes, int n_in,
                              void* d_out, int out_size, void* d_ws, size_t ws_size,
                              hipStream_t stream) {
  const float* x  = (const float*)d_in[0];  // [65536, 784]
  const float* cw = (const float*)d_in[1];  // [3, 3]
  const float* W  = (const float*)d_in[2];  // [676, 10]
  const float* b  = (const float*)d_in[3];  // [1, 10]
  float* out = (float*)d_out;               // [65536, 10]

  dim3 grid(N_IMGS / IMGS_PER_BLOCK);       // 2048
  dim3 block(BLOCK_THREADS);                // 64
  conv_relu_linear_wmma<<<grid, block, 0, stream>>>(x, cw, W, b, out);
}